// InfluenceMaximizationGNN_49460843381375
// MI455X (gfx1250) — compile-verified
//
#include <hip/hip_runtime.h>
#include <hip/hip_bf16.h>
#include <math.h>

typedef __attribute__((ext_vector_type(2))) float v2f;
typedef __attribute__((ext_vector_type(8))) float v8f;

#define HID 256

// ---------------- utility kernels ----------------

__global__ void k_zero(float* __restrict__ p, int n) {
    int i = blockIdx.x * blockDim.x + threadIdx.x;
    if (i < n) p[i] = 0.0f;
}

// degree count: deg[dst] += 1 per edge
__global__ void k_deg(const int* __restrict__ dst, float* __restrict__ deg, int E) {
    int e = blockIdx.x * blockDim.x + threadIdx.x;
    if (e < E) unsafeAtomicAdd(&deg[dst[e]], 1.0f);
}

// dis = rsqrt(deg+1), self_norm = 1/(deg+1)
__global__ void k_norm(const float* __restrict__ deg, float* __restrict__ dis,
                       float* __restrict__ sn, int N) {
    int i = blockIdx.x * blockDim.x + threadIdx.x;
    if (i < N) {
        float d = deg[i] + 1.0f;
        dis[i] = rsqrtf(d);
        sn[i]  = 1.0f / d;
    }
}

// edge_norm = dis[src]*dis[dst]
__global__ void k_edge_norm(const int* __restrict__ src, const int* __restrict__ dst,
                            const float* __restrict__ dis, float* __restrict__ en, int E) {
    int e = blockIdx.x * blockDim.x + threadIdx.x;
    if (e < E) en[e] = dis[src[e]] * dis[dst[e]];
}

// layer-1 transform: t[i,f] = x[i] * W1[f]   (W1 is 1 x H)
__global__ void k_l1_transform(const float* __restrict__ x, const float* __restrict__ W1,
                               float* __restrict__ t, int NH) {
    int idx = blockIdx.x * blockDim.x + threadIdx.x;
    if (idx < NH) {
        int i = idx >> 8;
        int f = idx & (HID - 1);
        t[idx] = x[i] * W1[f];
    }
}

// one 256-thread block per edge: a[dst,f] += t[src,f] * en[e]
__global__ void k_edge_agg(const float* __restrict__ t, float* __restrict__ a,
                           const int* __restrict__ src, const int* __restrict__ dst,
                           const float* __restrict__ en) {
    int e = blockIdx.x;
    int f = threadIdx.x;
    int s = src[e], d = dst[e];
    float w = en[e];
    unsafeAtomicAdd(&a[d * HID + f], t[s * HID + f] * w);
}

// t = relu(a + t*self_norm + b)   (in place into t)
__global__ void k_finalize(float* __restrict__ t, const float* __restrict__ a,
                           const float* __restrict__ sn, const float* __restrict__ b, int NH) {
    int idx = blockIdx.x * blockDim.x + threadIdx.x;
    if (idx < NH) {
        int i = idx >> 8;
        int f = idx & (HID - 1);
        float v = a[idx] + t[idx] * sn[i] + b[f];
        t[idx] = v > 0.0f ? v : 0.0f;
    }
}

// ---------------- WMMA f32 GEMM: C[M,256] = A[M,256] @ B[256,256] ----------------
// One wave computes a 16x16 C tile; K stepped by 4 via V_WMMA_F32_16X16X4_F32.
__global__ __launch_bounds__(256) void k_gemm_wmma(const float* __restrict__ A,
                                                   const float* __restrict__ B,
                                                   float* __restrict__ C, int M) {
    const int wid  = threadIdx.x >> 5;
    const int lane = threadIdx.x & 31;
    const int gw   = blockIdx.x * (blockDim.x >> 5) + wid;
    const int tiles_n = HID / 16;              // 16
    const int tm = gw / tiles_n;
    const int tn = gw % tiles_n;
    if (tm * 16 >= M) return;                  // wave-uniform: EXEC stays all-ones

    const int r = lane & 15;                   // m for A / n for B / col for C
    const int g = lane >> 4;                   // K-half selector (0 or 1)

    const float* __restrict__ arow = A + (size_t)(tm * 16 + r) * HID + g * 2;
    const float* __restrict__ bcol = B + (size_t)(g * 2) * HID + tn * 16 + r;

    v8f acc = {};
#pragma unroll 8
    for (int k = 0; k < HID; k += 4) {
        v2f av, bv;
        const float* ap = arow + k;
        av[0] = ap[0];
        av[1] = ap[1];
        const float* bp = bcol + (size_t)k * HID;
        bv[0] = bp[0];
        bv[1] = bp[HID];
        acc = __builtin_amdgcn_wmma_f32_16x16x4_f32(
            /*neg_a=*/false, av, /*neg_b=*/false, bv,
            /*c_mod=*/(short)0, acc, /*reuse_a=*/false, /*reuse_b=*/false);
    }

    const int row_base = tm * 16 + g * 8;
    const int col      = tn * 16 + r;
#pragma unroll
    for (int v = 0; v < 8; ++v)
        C[(size_t)(row_base + v) * HID + col] = acc[v];
}

// ---------------- pooling + head ----------------

__global__ void k_pool_sum(const float* __restrict__ t, const int* __restrict__ batch,
                           float* __restrict__ pooled, int NH) {
    int idx = blockIdx.x * blockDim.x + threadIdx.x;
    if (idx < NH) {
        int i = idx >> 8;
        int f = idx & (HID - 1);
        unsafeAtomicAdd(&pooled[batch[i] * HID + f], t[idx]);
    }
}

__global__ void k_pool_cnt(const int* __restrict__ batch, float* __restrict__ cnt, int N) {
    int i = blockIdx.x * blockDim.x + threadIdx.x;
    if (i < N) unsafeAtomicAdd(&cnt[batch[i]], 1.0f);
}

__global__ void k_pool_div(float* __restrict__ pooled, const float* __restrict__ cnt, int GH) {
    int idx = blockIdx.x * blockDim.x + threadIdx.x;
    if (idx < GH) {
        int gi = idx >> 8;
        pooled[idx] = pooled[idx] / fmaxf(cnt[gi], 1.0f);
    }
}

__global__ void k_bias_relu(float* __restrict__ h, const float* __restrict__ b, int GH) {
    int idx = blockIdx.x * blockDim.x + threadIdx.x;
    if (idx < GH) {
        int f = idx & (HID - 1);
        float v = h[idx] + b[f];
        h[idx] = v > 0.0f ? v : 0.0f;
    }
}

// out[g] = dot(h[g,:], w) + b0 ; one wave32 per graph
__global__ void k_fc2(const float* __restrict__ h, const float* __restrict__ w,
                      const float* __restrict__ b, float* __restrict__ out, int G) {
    int wid  = threadIdx.x >> 5;
    int lane = threadIdx.x & 31;
    int gidx = blockIdx.x * (blockDim.x >> 5) + wid;
    if (gidx >= G) return;
    float s = 0.0f;
    for (int f = lane; f < HID; f += 32)
        s += h[gidx * HID + f] * w[f];
#pragma unroll
    for (int off = 16; off > 0; off >>= 1)
        s += __shfl_xor(s, off, 32);
    if (lane == 0) out[gidx] = s + b[0];
}

// ---------------- launcher ----------------

extern "C" void kernel_launch(void* const* d_in, const int* in_sizes, int n_in,
                              void* d_out, int out_size, void* d_ws, size_t ws_size,
                              hipStream_t stream) {
    const float* x    = (const float*)d_in[0];
    const int*   ei   = (const int*)d_in[1];
    const int*   batch= (const int*)d_in[2];
    const float* W1   = (const float*)d_in[3];
    const float* b1   = (const float*)d_in[4];
    const float* W2   = (const float*)d_in[5];
    const float* b2   = (const float*)d_in[6];
    const float* W3   = (const float*)d_in[7];
    const float* b3   = (const float*)d_in[8];
    const float* W4   = (const float*)d_in[9];
    const float* b4   = (const float*)d_in[10];
    const float* fcW1 = (const float*)d_in[11];
    const float* fcb1 = (const float*)d_in[12];
    const float* fcW2 = (const float*)d_in[13];
    const float* fcb2 = (const float*)d_in[14];

    const int N = in_sizes[0];          // 20000
    const int E = in_sizes[1] / 2;      // 320000
    const int G = out_size;             // 256
    const int NH = N * HID;
    const int GH = G * HID;

    const int* src = ei;
    const int* dst = ei + E;

    // carve workspace (256B-aligned chunks)
    size_t off = 0;
    auto carve = [&](size_t bytes) -> float* {
        float* p = (float*)((char*)d_ws + off);
        off += (bytes + 255) & ~(size_t)255;
        return p;
    };
    float* t      = carve((size_t)NH * 4);   // transformed / layer output
    float* a      = carve((size_t)NH * 4);   // aggregation accumulator
    float* deg    = carve((size_t)N * 4);
    float* dis    = carve((size_t)N * 4);
    float* sn     = carve((size_t)N * 4);
    float* en     = carve((size_t)E * 4);
    float* pooled = carve((size_t)GH * 4);
    float* cnt    = carve((size_t)G * 4);
    float* fc1    = carve((size_t)GH * 4);
    (void)ws_size; (void)n_in;

    const int TB = 256;
    auto blocks = [](int n, int tb) { return (n + tb - 1) / tb; };

    // --- normalization terms ---
    k_zero<<<blocks(N, TB), TB, 0, stream>>>(deg, N);
    k_deg<<<blocks(E, TB), TB, 0, stream>>>(dst, deg, E);
    k_norm<<<blocks(N, TB), TB, 0, stream>>>(deg, dis, sn, N);
    k_edge_norm<<<blocks(E, TB), TB, 0, stream>>>(src, dst, dis, en, E);

    // --- layer 1 transform (outer product, W1 is 1xH) ---
    k_l1_transform<<<blocks(NH, TB), TB, 0, stream>>>(x, W1, t, NH);

    const float* Wnext[4] = { W2, W3, W4, nullptr };
    const float* bias[4]  = { b1, b2, b3, b4 };

    for (int l = 0; l < 4; ++l) {
        k_zero<<<blocks(NH, TB), TB, 0, stream>>>(a, NH);
        k_edge_agg<<<E, HID, 0, stream>>>(t, a, src, dst, en);
        k_finalize<<<blocks(NH, TB), TB, 0, stream>>>(t, a, sn, bias[l], NH);
        if (l < 3) {
            int tiles = (N / 16) * (HID / 16);
            k_gemm_wmma<<<blocks(tiles, 8), TB, 0, stream>>>(t, Wnext[l], a, N);
            float* tmp = t; t = a; a = tmp;   // transformed features now in t
        }
    }

    // --- global mean pool ---
    k_zero<<<blocks(GH, TB), TB, 0, stream>>>(pooled, GH);
    k_zero<<<blocks(G, TB), TB, 0, stream>>>(cnt, G);
    k_pool_sum<<<blocks(NH, TB), TB, 0, stream>>>(t, batch, pooled, NH);
    k_pool_cnt<<<blocks(N, TB), TB, 0, stream>>>(batch, cnt, N);
    k_pool_div<<<blocks(GH, TB), TB, 0, stream>>>(pooled, cnt, GH);

    // --- head: fc1 (WMMA GEMM) + relu, fc2 (wave dot) ---
    {
        int tiles = (G / 16) * (HID / 16);
        k_gemm_wmma<<<blocks(tiles, 8), TB, 0, stream>>>(pooled, fcW1, fc1, G);
    }
    k_bias_relu<<<blocks(GH, TB), TB, 0, stream>>>(fc1, fcb1, GH);
    k_fc2<<<blocks(G, 8), TB, 0, stream>>>(fc1, fcW2, fcb2, (float*)d_out, G);
}